// PhysicsConstraintValidator_60404420051240
// MI455X (gfx1250) — compile-verified
//
#include <hip/hip_runtime.h>
#include <math.h>

typedef __attribute__((ext_vector_type(2))) float v2f;
typedef __attribute__((ext_vector_type(8))) float v8f;

#define BATCH 16
#define NPTS 2048
#define NTILES 128                         // NPTS / 16
#define PAIRS_PER_BATCH 8256               // NTILES*(NTILES+1)/2
#define TOTAL_TILE_PAIRS (BATCH * PAIRS_PER_BATCH)  // 132096
#define WAVES_PER_BLOCK 8
#define CAUS_BLOCKS (TOTAL_TILE_PAIRS / WAVES_PER_BLOCK)  // 16512, exact
#define TOTAL_PAIRS_F 2096128.0f           // N*(N-1)/2

__device__ __forceinline__ int tri_off(int i) {
    // start offset of row i in row-major upper triangle (incl. diagonal), T=128
    return i * NTILES - (i * (i - 1)) / 2;
}

// ---------------------------------------------------------------------------
// Kernel 1: per-batch reductions (energy / momentum / lorentz partials).
// Also zeroes the causality counters (runs before the WMMA kernel on stream).
// ---------------------------------------------------------------------------
__global__ void __launch_bounds__(256)
reduce_kernel(const float4* __restrict__ ev, const float4* __restrict__ pr,
              float* __restrict__ partials, unsigned int* __restrict__ counts) {
    const int b = blockIdx.x;
    if (threadIdx.x == 0) counts[b] = 0u;

    float acc[9];
    #pragma unroll
    for (int k = 0; k < 9; ++k) acc[k] = 0.0f;

    for (int i = threadIdx.x; i < NPTS; i += 256) {
        float4 e = ev[b * NPTS + i];
        float4 p = pr[b * NPTS + i];
        acc[0] += e.x;  acc[1] += p.x;                    // E_i, E_f
        acc[2] += e.y;  acc[3] += e.z;  acc[4] += e.w;    // p_i
        acc[5] += p.y;  acc[6] += p.z;  acc[7] += p.w;    // p_f
        float mi2 = e.x * e.x - (e.y * e.y + e.z * e.z + e.w * e.w);
        float mf2 = p.x * p.x - (p.y * p.y + p.z * p.z + p.w * p.w);
        float mi = sqrtf(fmaxf(mi2, 0.0f));
        float mf = sqrtf(fmaxf(mf2, 0.0f));
        acc[8] += fabsf(mf - mi) / (mi + 1e-6f);          // lorentz sum
    }

    // wave32 shuffle reduction
    #pragma unroll
    for (int k = 0; k < 9; ++k)
        for (int off = 16; off > 0; off >>= 1)
            acc[k] += __shfl_down(acc[k], off, 32);

    __shared__ float sbuf[8][9];
    const int wave = threadIdx.x >> 5;
    const int lane = threadIdx.x & 31;
    if (lane == 0) {
        #pragma unroll
        for (int k = 0; k < 9; ++k) sbuf[wave][k] = acc[k];
    }
    __syncthreads();
    if (threadIdx.x == 0) {
        #pragma unroll
        for (int k = 0; k < 9; ++k) {
            float s = 0.0f;
            #pragma unroll
            for (int w = 0; w < 8; ++w) s += sbuf[w][k];
            partials[b * 9 + k] = s;
        }
    }
}

// ---------------------------------------------------------------------------
// Kernel 2: causality pair counting via V_WMMA_F32_16X16X4_F32.
// One wave per (batch, upper-tri 16x16 tile pair). K=4 is the 4-vector.
// M_nm = t_n*t_m - x_n.x_m computed by WMMA with metric signs folded into B.
// Violation: s_n + s_m < 2*M_nm, with n < m (strict upper triangle).
// ---------------------------------------------------------------------------
__global__ void __launch_bounds__(256)
caus_kernel(const float* __restrict__ coords, unsigned int* __restrict__ counts) {
    const int lane = threadIdx.x & 31;
    const int wid  = blockIdx.x * WAVES_PER_BLOCK + (threadIdx.x >> 5);
    const int b = wid / PAIRS_PER_BATCH;
    const int p = wid - b * PAIRS_PER_BATCH;

    // invert triangular index: p -> (ti, tj), ti <= tj
    int ti = (int)((257.0f - sqrtf(66049.0f - 8.0f * (float)p)) * 0.5f);
    ti = ti < 0 ? 0 : (ti > NTILES - 1 ? NTILES - 1 : ti);
    while (ti < NTILES - 1 && tri_off(ti + 1) <= p) ++ti;
    while (ti > 0 && tri_off(ti) > p) --ti;
    const int tj = ti + (p - tri_off(ti));

    const float* cb = coords + (size_t)b * NPTS * 4;
    const int r0   = lane & 15;   // row (A) / column (B,D) within tile
    const int half = lane >> 4;   // selects K=0,1 vs K=2,3 per ISA f32 layout

    // A fragment: i-tile rows, lane holds components (2*half, 2*half+1)
    const float* pa = cb + (ti * 16 + r0) * 4 + half * 2;
    v2f a;
    a.x = pa[0];
    a.y = pa[1];

    // B fragment: j-tile columns with Minkowski metric (+,-,-,-)
    const float* pb = cb + (tj * 16 + r0) * 4 + half * 2;
    float b0 = pb[0], b1 = pb[1];
    v2f bm;
    bm.x = half ? -b0 : b0;   // t keeps +, x/y/z get -
    bm.y = -b1;

    v8f c = {0.f, 0.f, 0.f, 0.f, 0.f, 0.f, 0.f, 0.f};
    // 8 args: (neg_a, A, neg_b, B, c_mod, C, reuse_a, reuse_b)
    v8f d = __builtin_amdgcn_wmma_f32_16x16x4_f32(
        false, a, false, bm, (short)0, c, false, false);

    // Minkowski norms s = t^2 - |x|^2 (coords are L2-resident; broadcast loads)
    const float4* cb4 = (const float4*)cb;
    const int mcol = tj * 16 + r0;          // D column index for this lane
    float4 q = cb4[mcol];
    const float s_col = q.x * q.x - (q.y * q.y + q.z * q.z + q.w * q.w);

    unsigned int cnt = 0;
    #pragma unroll
    for (int g = 0; g < 8; ++g) {
        const int nrow = ti * 16 + half * 8 + g;   // D row per ISA C/D layout
        float4 r = cb4[nrow];
        float s_row = r.x * r.x - (r.y * r.y + r.z * r.z + r.w * r.w);
        bool viol = (s_row + s_col < 2.0f * d[g]) && (nrow < mcol);
        unsigned long long bal = __ballot(viol);
        cnt += (unsigned int)__popcll(bal);
    }
    if (lane == 0) atomicAdd(&counts[b], cnt);
}

// ---------------------------------------------------------------------------
// Kernel 3: combine partials + counts into the (16,5) output.
// ---------------------------------------------------------------------------
__global__ void finalize_kernel(const float* __restrict__ partials,
                                const unsigned int* __restrict__ counts,
                                float* __restrict__ out) {
    const int b = threadIdx.x;
    if (b >= BATCH) return;
    const float* q = partials + b * 9;
    const float ei = q[0], ef = q[1];
    const float pix = q[2], piy = q[3], piz = q[4];
    const float pfx = q[5], pfy = q[6], pfz = q[7];

    const float energy_viol = fabsf(ef - ei) / ei;   // matches ref (no |E_i|)
    const float dx = pfx - pix, dy = pfy - piy, dz = pfz - piz;
    const float mom_viol =
        sqrtf(dx * dx + dy * dy + dz * dz) /
        sqrtf(pix * pix + piy * piy + piz * piz);
    const float lorentz_viol = q[8] / (float)NPTS;
    const float caus_viol = (float)counts[b] / TOTAL_PAIRS_F;
    const float overall =
        1.0f - (energy_viol + mom_viol + lorentz_viol + caus_viol) * 0.25f;

    out[b * 5 + 0] = energy_viol;
    out[b * 5 + 1] = mom_viol;
    out[b * 5 + 2] = lorentz_viol;
    out[b * 5 + 3] = caus_viol;
    out[b * 5 + 4] = overall;
}

extern "C" void kernel_launch(void* const* d_in, const int* in_sizes, int n_in,
                              void* d_out, int out_size, void* d_ws, size_t ws_size,
                              hipStream_t stream) {
    const float* event  = (const float*)d_in[0];
    const float* pred   = (const float*)d_in[1];
    const float* coords = (const float*)d_in[2];
    float* out = (float*)d_out;

    unsigned int* counts = (unsigned int*)d_ws;          // 16 u32
    float* partials = (float*)d_ws + 16;                 // 16*9 floats

    reduce_kernel<<<BATCH, 256, 0, stream>>>(
        (const float4*)event, (const float4*)pred, partials, counts);
    caus_kernel<<<CAUS_BLOCKS, 256, 0, stream>>>(coords, counts);
    finalize_kernel<<<1, 32, 0, stream>>>(partials, counts, out);
}